// Upsample_16484084483533
// MI455X (gfx1250) — compile-verified
//
#include <hip/hip_runtime.h>
#include <hip/hip_bf16.h>

// Upsample2x with FIR [1,3,3,1] (factor=2): polyphase weights [0.25, 0.75] per dim.
// Input:  (8,128,128,128) f32  -> 1024 images of 128x128
// Output: (8,128,256,256) f32  -> 1024 images of 256x256
//
// Bandwidth-bound (~320 MB @ 23.3 TB/s => ~14us floor). Strategy:
//   - stage 10-row input tile (8 rows + 2 halo) into LDS with
//     GLOBAL_LOAD_ASYNC_TO_LDS_B128 (CDNA5 async copy, ASYNCcnt-tracked)
//   - compute separable polyphase FIR from LDS
//   - 128-bit coalesced NON-TEMPORAL stores (output is write-once streaming;
//     input fits in the 192 MB L2 and should not be evicted by the out stream)

#define IN_H   128
#define IN_W   128
#define OUT_W  256
#define NIMG   1024   // 8 batches * 128 channels
#define ROWS   8      // input rows per block -> 16 output rows

typedef float v4f __attribute__((ext_vector_type(4)));

__global__ __launch_bounds__(256)
void upfir2x_kernel(const float* __restrict__ x, float* __restrict__ out)
{
    __shared__ float tile[(ROWS + 2) * IN_W];   // 10 * 128 floats = 5 KB

    const int t  = threadIdx.x;
    const int bc = blockIdx.y;                  // image index 0..1023
    const int i0 = blockIdx.x * ROWS;           // first input row of this tile

    // ---- Stage rows (i0-1 .. i0+ROWS) into LDS via async global->LDS copies.
    // 320 b128 slots, 256 threads: slot t and (for t<64) slot t+256.
    // Halo rows outside [0,128) are zero-filled with DS stores instead.
    #pragma unroll
    for (int s = t; s < (ROWS + 2) * (IN_W / 4); s += 256) {
        const int row  = s >> 5;                // 0..9 (LDS tile row)
        const int col4 = s & 31;                // 0..31 (float4 within row)
        const int g    = i0 - 1 + row;          // global input row
        float* ldsp = &tile[row * IN_W + col4 * 4];
        if ((unsigned)g < (unsigned)IN_H) {
            // low 32 bits of the generic pointer == wave-relative LDS byte offset
            const unsigned lda = (unsigned)(uintptr_t)ldsp;
            const unsigned long long ga = (unsigned long long)(uintptr_t)
                (x + ((size_t)bc * IN_H + (size_t)g) * IN_W + col4 * 4);
            asm volatile("global_load_async_to_lds_b128 %0, %1, off"
                         :: "v"(lda), "v"(ga)
                         : "memory");
        } else {
            ldsp[0] = 0.0f; ldsp[1] = 0.0f; ldsp[2] = 0.0f; ldsp[3] = 0.0f;
        }
    }
    // Each wave drains its own async copies, then workgroup barrier.
    asm volatile("s_wait_asynccnt 0x0" ::: "memory");
    __syncthreads();

    // ---- Compute: thread t handles input rows i = i0 + r + 4k (k = 0,1),
    //      input cols (2c, 2c+1) -> output rows (2i, 2i+1), cols (4c .. 4c+3).
    const int r = t >> 6;                       // 0..3
    const int c = t & 63;                       // 0..63
    const int j0 = 2 * c;
    const bool lo = (c == 0);
    const bool hi = (c == 63);
    const float A = 0.25f, B = 0.75f;

    const size_t obase = (size_t)bc * OUT_W * OUT_W;

    #pragma unroll
    for (int k = 0; k < 2; ++k) {
        const int rr = r + 4 * k;               // 0..7: row within tile
        const float* rt = &tile[(rr + 0) * IN_W]; // global row i-1
        const float* rm = &tile[(rr + 1) * IN_W]; // global row i
        const float* rb = &tile[(rr + 2) * IN_W]; // global row i+1

        const float tm1 = lo ? 0.0f : rt[j0 - 1];
        const float t0  = rt[j0];
        const float t1  = rt[j0 + 1];
        const float t2  = hi ? 0.0f : rt[j0 + 2];

        const float mm1 = lo ? 0.0f : rm[j0 - 1];
        const float m0  = rm[j0];
        const float m1  = rm[j0 + 1];
        const float m2  = hi ? 0.0f : rm[j0 + 2];

        const float bm1 = lo ? 0.0f : rb[j0 - 1];
        const float b0  = rb[j0];
        const float b1  = rb[j0 + 1];
        const float b2  = hi ? 0.0f : rb[j0 + 2];

        // Horizontal polyphase (4 output columns) for each of the 3 rows.
        const float ht0 = fmaf(A, tm1, B * t0);
        const float ht1 = fmaf(B, t0,  A * t1);
        const float ht2 = fmaf(A, t0,  B * t1);
        const float ht3 = fmaf(B, t1,  A * t2);

        const float hm0 = fmaf(A, mm1, B * m0);
        const float hm1 = fmaf(B, m0,  A * m1);
        const float hm2 = fmaf(A, m0,  B * m1);
        const float hm3 = fmaf(B, m1,  A * m2);

        const float hb0 = fmaf(A, bm1, B * b0);
        const float hb1 = fmaf(B, b0,  A * b1);
        const float hb2 = fmaf(A, b0,  B * b1);
        const float hb3 = fmaf(B, b1,  A * b2);

        // Vertical polyphase: even output row uses (i-1, i), odd uses (i, i+1).
        v4f e, o;
        e.x = fmaf(A, ht0, B * hm0);
        e.y = fmaf(A, ht1, B * hm1);
        e.z = fmaf(A, ht2, B * hm2);
        e.w = fmaf(A, ht3, B * hm3);
        o.x = fmaf(B, hm0, A * hb0);
        o.y = fmaf(B, hm1, A * hb1);
        o.z = fmaf(B, hm2, A * hb2);
        o.w = fmaf(B, hm3, A * hb3);

        const int i = i0 + rr;                  // global input row
        float* orow = out + obase + (size_t)(2 * i) * OUT_W + 4 * c;
        __builtin_nontemporal_store(e, (v4f*)orow);            // row 2i
        __builtin_nontemporal_store(o, (v4f*)(orow + OUT_W));  // row 2i+1
    }
}

extern "C" void kernel_launch(void* const* d_in, const int* in_sizes, int n_in,
                              void* d_out, int out_size, void* d_ws, size_t ws_size,
                              hipStream_t stream)
{
    const float* x = (const float*)d_in[0];
    float* out     = (float*)d_out;

    dim3 grid(IN_H / ROWS, NIMG);   // 16 x 1024 blocks
    dim3 block(256);                // 8 wave32 waves
    upfir2x_kernel<<<grid, block, 0, stream>>>(x, out);
}